// FreeEnergy_34686155883200
// MI455X (gfx1250) — compile-verified
//
#include <hip/hip_runtime.h>
#include <math.h>

#define N_SAMPLES 8000
#define NODES 64
#define NM1 63
#define NM2 62
#define LDX 66              // LDS row stride in dwords: 64 (padded K) + 2 anti-bank-conflict
#define LAMBDA_MAX 0.06f
#define NWAVES 5            // 5 waves x 2 tiles = 10 upper-triangular 16x16 tiles

typedef __attribute__((ext_vector_type(2))) float v2f;
typedef __attribute__((ext_vector_type(8))) float v8f;

__device__ __forceinline__ float wave_sum32(float v) {
#pragma unroll
  for (int off = 16; off > 0; off >>= 1) v += __shfl_xor(v, off, 32);
  return v;
}

__device__ __forceinline__ float softplusf(float x) {
  return fmaxf(x, 0.f) + log1pf(expf(-fabsf(x)));
}

// enumerate upper triangle of the 4x4 tile grid: t=0..9 -> (I<=J)
__device__ __forceinline__ void tile_ij(int t, int& I, int& J) {
  if (t < 4)      { I = 0; J = t; }
  else if (t < 7) { I = 1; J = t - 3; }
  else if (t < 9) { I = 2; J = t - 5; }
  else            { I = 3; J = 3; }
}

__global__ __launch_bounds__(32 * NWAVES) void fe_traj_ll_kernel(
    const float* __restrict__ raw_d, const float* __restrict__ raw_c,
    const float* __restrict__ raw_a, const float* __restrict__ raw_l,
    const int* __restrict__ trials_all, const int* __restrict__ trials_correct,
    float* __restrict__ per_sample_ll) {
  __shared__ float X[NODES * LDX];   // trajectory, row-major, col 63 zero-padded
  __shared__ float SQ[NODES];        // squared norms of rows
  __shared__ float red[NWAVES];

  const int s = blockIdx.x;
  const int tid = threadIdx.x;
  const int wave = tid >> 5;
  const int lane = tid & 31;

  // ---------- Phase 1: sequential trajectory recurrence (wave 0 only) ----------
  if (wave == 0) {
    const int d0 = lane;             // dims 0..31
    const int d1 = lane + 32;        // dims 32..62 (lane 31 slot invalid)
    const bool ok1 = (d1 < NM1);

    // zero the K-pad column (col 63) for all 64 rows
    X[(2 * lane) * LDX + 63] = 0.f;
    X[(2 * lane + 1) * LDX + 63] = 0.f;

    float v0 = (d0 == 0) ? 1.f : 0.f;   // v1 = e0
    float v1 = 0.f;
    float x0 = softplusf(raw_d[(size_t)s * NM1]) * v0;
    float x1 = 0.f;

    X[0 * LDX + d0] = 0.f;                   // x row 0 = zeros
    if (ok1) X[0 * LDX + d1] = 0.f;
    X[1 * LDX + d0] = x0;                    // x row 1 = d0 * e0
    if (ok1) X[1 * LDX + d1] = 0.f;

    const float* arow = raw_a + (size_t)s * NM2 * NM1;
    for (int i = 0; i < NM2; ++i) {
      float a0 = arow[i * NM1 + d0];
      float a1 = ok1 ? arow[i * NM1 + d1] : 0.f;
      // Gram-Schmidt against previous heading
      float proj = wave_sum32(a0 * v0 + a1 * v1);
      a0 -= proj * v0;
      a1 -= proj * v1;
      float n2 = wave_sum32(a0 * a0 + a1 * a1);
      float inv = 1.f / sqrtf(n2);
      // rotate heading by angle c, advance by distance d
      float ci = raw_c[(size_t)s * NM2 + i];
      float cs = cosf(ci), sn = sinf(ci);
      v0 = cs * v0 + sn * a0 * inv;
      v1 = cs * v1 + sn * a1 * inv;
      float di = softplusf(raw_d[(size_t)s * NM1 + i + 1]);
      x0 += di * v0;
      x1 += di * v1;
      X[(i + 2) * LDX + d0] = x0;
      if (ok1) X[(i + 2) * LDX + d1] = x1;
    }
  }
  __syncthreads();

  // ---------- Phase 2: per-row squared norms ----------
  if (tid < NODES) {
    float acc = 0.f;
    for (int k = 0; k < NM1; ++k) { float t = X[tid * LDX + k]; acc += t * t; }
    SQ[tid] = acc;
  }
  __syncthreads();

  // ---------- Phase 3: gram via FP32 WMMA + symmetric likelihood ----------
  const float lam = LAMBDA_MAX * 0.5f * (1.f + erff(raw_l[s] * 0.70710678118654752f));
  const float one_m2l = 1.f - 2.f * lam;

  const int nidx = lane & 15;      // M (for A) / N (for B, C/D col)
  const int hi = lane >> 4;        // lane half selects K pair / C row offset
  const int kb = hi * 2;

  float ll = 0.f;
#pragma unroll
  for (int t = 0; t < 2; ++t) {
    int I, J;
    tile_ij(wave * 2 + t, I, J);   // upper-triangular tile (I <= J)
    const int mbase = (I * 16 + nidx) * LDX;
    const int nbase = (J * 16 + nidx) * LDX;
    v8f acc = {};
#pragma unroll
    for (int kk = 0; kk < 64; kk += 4) {
      // A(16x4): lane=M, VGPR pair = K {kb, kb+1};  B(4x16): identical indexing of X^T chunk
      v2f a = *(const v2f*)&X[mbase + kk + kb];
      v2f b = *(const v2f*)&X[nbase + kk + kb];
      acc = __builtin_amdgcn_wmma_f32_16x16x4_f32(false, a, false, b, (short)0, acc,
                                                  false, false);
    }
    // C/D layout: VGPR e holds row M = e + 8*hi, col N = lane&15
#pragma unroll
    for (int e = 0; e < 8; ++e) {
      const int gi = I * 16 + e + 8 * hi;
      const int gj = J * 16 + nidx;
      float pd2 = fmaxf(SQ[gi] + SQ[gj] - 2.f * acc[e], 0.f);
      float pd = sqrtf(pd2);
      float z1 = 0.5f * (1.f + erff(pd * 0.5f));                   // Phi(pd/sqrt2)
      float z2 = 0.5f * (1.f + erff(pd * 0.35355339059327373f));   // Phi(pd/2)
      float paxb = z1 * z2 + (1.f - z1) * (1.f - z2);
      float p = one_m2l * paxb + lam;
      float lp = logf(p);
      float l1mp = log1pf(-p);
      // (gi, gj) contribution
      float nc = (float)trials_correct[gi * 64 + gj];
      float na = (float)trials_all[gi * 64 + gj];
      ll += nc * lp + (na - nc) * l1mp;
      // symmetric (gj, gi) contribution reuses the same p (distance symmetric)
      if (I != J) {
        float ncT = (float)trials_correct[gj * 64 + gi];
        float naT = (float)trials_all[gj * 64 + gi];
        ll += ncT * lp + (naT - ncT) * l1mp;
      }
    }
  }

  // deterministic block reduction
  ll = wave_sum32(ll);
  if (lane == 0) red[wave] = ll;
  __syncthreads();
  if (tid == 0) {
    float acc = 0.f;
#pragma unroll
    for (int w = 0; w < NWAVES; ++w) acc += red[w];
    per_sample_ll[s] = acc;
  }
}

__global__ __launch_bounds__(256) void fe_reduce_mean_kernel(
    const float* __restrict__ per_sample_ll, float* __restrict__ out) {
  __shared__ float sh[256];
  float acc = 0.f;
  for (int i = threadIdx.x; i < N_SAMPLES; i += 256) acc += per_sample_ll[i];
  sh[threadIdx.x] = acc;
  __syncthreads();
  for (int st = 128; st > 0; st >>= 1) {
    if (threadIdx.x < st) sh[threadIdx.x] += sh[threadIdx.x + st];
    __syncthreads();
  }
  if (threadIdx.x == 0) out[0] = sh[0] * (1.f / (float)N_SAMPLES);
}

extern "C" void kernel_launch(void* const* d_in, const int* in_sizes, int n_in,
                              void* d_out, int out_size, void* d_ws, size_t ws_size,
                              hipStream_t stream) {
  const float* raw_d = (const float*)d_in[0];
  const float* raw_c = (const float*)d_in[1];
  const float* raw_a = (const float*)d_in[2];
  const float* raw_l = (const float*)d_in[3];
  const int* trials_all = (const int*)d_in[4];
  const int* trials_correct = (const int*)d_in[5];
  float* per_sample_ll = (float*)d_ws;   // 8000 floats of scratch
  float* out = (float*)d_out;

  fe_traj_ll_kernel<<<N_SAMPLES, 32 * NWAVES, 0, stream>>>(
      raw_d, raw_c, raw_a, raw_l, trials_all, trials_correct, per_sample_ll);
  fe_reduce_mean_kernel<<<1, 256, 0, stream>>>(per_sample_ll, out);
}